// FastSAM3DPromptEncoder_74354473828895
// MI455X (gfx1250) — compile-verified
//
#include <hip/hip_runtime.h>

// FastSAM3D prompt encoder: clamp->gather(pos_embed, channel stride 1MB)->+type_emb.
// Memory-bound gather; roofline floor ~0.4GB / 23.3TB/s ~= 17-25us on MI455X.
// CDNA5-specific paths used: async global->LDS staging (ASYNCcnt),
// non-temporal (TH=NT) output stores to preserve 192MB L2 for the gather table,
// wave-uniform readfirstlane scalarization for saddr+ioffset gather addressing.

typedef float v4f __attribute__((ext_vector_type(4)));

typedef __attribute__((address_space(1))) int gint_t;   // global (AS1)
typedef __attribute__((address_space(3))) int lint_t;   // LDS (AS3)

#define GRID_D     64
#define CH         256
#define CH_STRIDE  (GRID_D * GRID_D * GRID_D)   // 262144 floats between channels
#define SUBS       4                            // concurrent point subgroups (64 thr each)
#define ITERS      2                            // unrolled point iterations per block
#define PPB        (SUBS * ITERS)               // 8 points per block

__global__ __launch_bounds__(256) void fastsam3d_prompt_encoder(
    const float* __restrict__ points,     // [B*N, 3]
    const int*   __restrict__ labels,     // [B*N]
    const float* __restrict__ pos_embed,  // [CH, 64,64,64]
    const float* __restrict__ point_emb,  // [2, CH]
    float*       __restrict__ out,        // [B*N, CH]
    int n_points)
{
    const int tid   = threadIdx.x;
    const int pbase = blockIdx.x * PPB;

    // [0..23] = 8 points x 3 coords (float), [24..31] = 8 labels (raw bit copy)
    __shared__ float s_stage[32];

    // ---- Stage this block's point data into LDS (CDNA5 async DMA path) ----
    if (tid < 32) {
        const bool is_pt = (tid < 24);
        const void* src = is_pt ? (const void*)(points + (size_t)pbase * 3 + tid)
                                : (const void*)(labels + pbase + (tid - 24));
        const bool valid = is_pt ? ((size_t)pbase * 3 + tid < (size_t)n_points * 3)
                                 : (pbase + (tid - 24) < n_points);
#if __has_builtin(__builtin_amdgcn_global_load_async_to_lds_b32)
        if (valid) {
            __builtin_amdgcn_global_load_async_to_lds_b32(
                (gint_t*)src,                    // global source (AS1)
                (lint_t*)&s_stage[tid],          // LDS destination (AS3)
                /*imm offset*/ 0, /*cpol*/ 0);
        }
#if __has_builtin(__builtin_amdgcn_s_wait_asynccnt)
        __builtin_amdgcn_s_wait_asynccnt(0);
#else
        asm volatile("s_wait_asynccnt 0" ::: "memory");
#endif
#else
        if (valid) {
            s_stage[tid] = is_pt ? *(const float*)src
                                 : __int_as_float(*(const int*)src);
        }
#endif
    }
    __syncthreads();

    // ---- Register-cache both type embeddings for this thread's 4 channels ----
    const int c0  = (tid & 63) * 4;                      // channel group, 16B aligned
    const v4f te0 = *(const v4f*)(point_emb + c0);       // label == 0
    const v4f te1 = *(const v4f*)(point_emb + CH + c0);  // label == 1

    const int sub = tid >> 6;   // wave-uniform subgroup id 0..3

#pragma unroll
    for (int i = 0; i < ITERS; ++i) {
        const int q = i * SUBS + sub;   // point slot within block, wave-uniform
        const int p = pbase + q;        // global point id
        if (p >= n_points) continue;    // uniform branch per wave

        // Broadcast point data from LDS (same address across the wave)
        const float pz = s_stage[q * 3 + 0];
        const float py = s_stage[q * 3 + 1];
        const float px = s_stage[q * 3 + 2];
        int lab = __float_as_int(s_stage[24 + q]);

        // int() truncation (coords >= 0) then clamp to [0, 63] — matches reference
        const int zi = min(max((int)pz, 0), GRID_D - 1);
        const int yi = min(max((int)py, 0), GRID_D - 1);
        const int xi = min(max((int)px, 0), GRID_D - 1);

        // Scalarize: gather addresses become saddr + small imm offsets (0..3MB)
        const int base = __builtin_amdgcn_readfirstlane(
            zi * (GRID_D * GRID_D) + yi * GRID_D + xi);
        lab = __builtin_amdgcn_readfirstlane(lab);

        // Channel-strided gather: 4 loads, 1MB apart (keep RT hint: L2 reuse)
        const float* g = pos_embed + (size_t)c0 * CH_STRIDE + base;
        const float v0 = g[0];
        const float v1 = g[1 * CH_STRIDE];
        const float v2 = g[2 * CH_STRIDE];
        const float v3 = g[3 * CH_STRIDE];

        const v4f te = lab ? te1 : te0;
        v4f r;
        r.x = v0 + te.x;
        r.y = v1 + te.y;
        r.z = v2 + te.z;
        r.w = v3 + te.w;

        // Streaming output: non-temporal store so the 192MB L2 keeps pos_embed
        __builtin_nontemporal_store(r, (v4f*)(out + (size_t)p * CH + c0));
    }
}

extern "C" void kernel_launch(void* const* d_in, const int* in_sizes, int n_in,
                              void* d_out, int out_size, void* d_ws, size_t ws_size,
                              hipStream_t stream) {
    const float* points    = (const float*)d_in[0];  // [B, N, 3] f32
    const int*   labels    = (const int*)  d_in[1];  // [B, N]    i32
    const float* pos_embed = (const float*)d_in[2];  // [1, 256, 64,64,64] f32
    const float* point_emb = (const float*)d_in[3];  // [2, 256]  f32
    float*       out       = (float*)d_out;          // [B, N, 256] f32

    const int n_points = in_sizes[1];                // B*N = 131072
    const int blocks   = (n_points + PPB - 1) / PPB; // 16384 blocks x 256 threads

    fastsam3d_prompt_encoder<<<blocks, 256, 0, stream>>>(
        points, labels, pos_embed, point_emb, out, n_points);
}